// DynamicGraphEmbedding_17660905521895
// MI455X (gfx1250) — compile-verified
//
#include <hip/hip_runtime.h>
#include <math.h>

// ---------------- problem constants ----------------
#define B_   8
#define N_   2048
#define W_   128
#define D_   64
#define K_   20
#define BN_  (B_ * N_)      // 16384
#define EPS_ 1e-5f
#define SLOPE_ 0.2f
#define NEG_BIG (-3.402823466e38f)

typedef __attribute__((ext_vector_type(2))) float v2f;
typedef __attribute__((ext_vector_type(8))) float v8f;

// Full-precision WMMA: D(16x16,f32) = A(16x4,f32) * B(4x16,f32) + C
// A layout: lanes 0-15 -> M=lane, {K=k,k+1}; lanes 16-31 -> M=lane-16, {K=k+2,k+3}
// B layout: lanes 0-15 -> N=lane, {K=k,k+1}; lanes 16-31 -> N=lane-16, {K=k+2,k+3}
// C/D: VGPR r, lane l -> M = r + 8*(l>=16), N = l&15
static __device__ inline v8f wmma_f32x4(v2f a, v2f b, v8f c) {
    return __builtin_amdgcn_wmma_f32_16x16x4_f32(
        /*neg_a=*/false, a, /*neg_b=*/false, b,
        /*c_mod=*/(short)0, c, /*reuse_a=*/false, /*reuse_b=*/false);
}

// ---------------- K1: depthwise TCN + residual ----------------
__global__ void tcn_kernel(const float* __restrict__ x,
                           const float* __restrict__ w1, const float* __restrict__ b1,
                           const float* __restrict__ s1, const float* __restrict__ o1,
                           const float* __restrict__ w2, const float* __restrict__ b2,
                           const float* __restrict__ s2, const float* __restrict__ o2,
                           const float* __restrict__ w3, const float* __restrict__ b3,
                           const float* __restrict__ s3, const float* __restrict__ o3,
                           float* __restrict__ xp) {
    __shared__ float xs[W_];
    const int row = blockIdx.x;          // b*N + n
    const int n = row & (N_ - 1);
    const int t = threadIdx.x;
    const float* xr = x + (size_t)row * W_;
    xs[t] = xr[t];
    __syncthreads();
    const float rs = rsqrtf(1.0f + EPS_);

    float acc = 0.0f;
    // branch 1: k=3, pad=1
    {
        float y = 0.0f;
#pragma unroll
        for (int k = 0; k < 3; ++k) {
            int j = t - 1 + k;
            float xv = (j >= 0 && j < W_) ? xs[j] : 0.0f;
            y += xv * w1[n * 3 + k];
        }
        y = (y + b1[n]) * (s1[n] * rs) + o1[n];
        acc += fmaxf(y, 0.0f);
    }
    // branch 2: k=5, pad=2
    {
        float y = 0.0f;
#pragma unroll
        for (int k = 0; k < 5; ++k) {
            int j = t - 2 + k;
            float xv = (j >= 0 && j < W_) ? xs[j] : 0.0f;
            y += xv * w2[n * 5 + k];
        }
        y = (y + b2[n]) * (s2[n] * rs) + o2[n];
        acc += fmaxf(y, 0.0f);
    }
    // branch 3: k=7, pad=3
    {
        float y = 0.0f;
#pragma unroll
        for (int k = 0; k < 7; ++k) {
            int j = t - 3 + k;
            float xv = (j >= 0 && j < W_) ? xs[j] : 0.0f;
            y += xv * w3[n * 7 + k];
        }
        y = (y + b3[n]) * (s3[n] * rs) + o3[n];
        acc += fmaxf(y, 0.0f);
    }
    xp[(size_t)row * W_ + t] = xs[t] + acc * (1.0f / 3.0f);
}

// ---------------- K2: row-normalize embedding ----------------
__global__ void norm_emb_kernel(const float* __restrict__ emb, float* __restrict__ e) {
    __shared__ float red[D_];
    const int r = blockIdx.x;
    const int t = threadIdx.x;     // 64 threads
    float v = emb[(size_t)r * D_ + t];
    red[t] = v * v;
    __syncthreads();
    for (int s = 32; s > 0; s >>= 1) {
        if (t < s) red[t] += red[t + s];
        __syncthreads();
    }
    float nrm = sqrtf(red[0]) + 1e-12f;
    e[(size_t)r * D_ + t] = v / nrm;
}

// ---------------- K3: sim = e·e^T (WMMA) + iterative top-K ----------------
// Block: 256 threads (8 waves) handles 16 rows. sim row buffer in dynamic LDS:
// 16 * 2048 floats = 128 KB (within the 320 KB/WGP budget).
__global__ void sim_topk_kernel(const float* __restrict__ e,
                                float* __restrict__ tvals, int* __restrict__ tidx) {
    extern __shared__ float sim[];   // [16][N_]
    const int i0 = blockIdx.x * 16;
    const int lane = threadIdx.x & 31;
    const int wave = threadIdx.x >> 5;       // 0..7
    const int rsel = lane & 15;
    const int khalf = (lane >> 4) << 1;      // 0 or 2

    // Preload A fragments for this block's 16 i-rows (K=64 -> 16 steps of 4)
    v2f afr[16];
#pragma unroll
    for (int ks = 0; ks < 16; ++ks) {
        const int kk = ks * 4 + khalf;
        const float* row = e + (size_t)(i0 + rsel) * D_ + kk;
        afr[ks].x = row[0];
        afr[ks].y = row[1];
    }
    // Each wave computes 16 of the 128 column tiles
    for (int jt = wave; jt < N_ / 16; jt += 8) {
        const int j0 = jt * 16;
        v8f c = {};
#pragma unroll
        for (int ks = 0; ks < 16; ++ks) {
            const int kk = ks * 4 + khalf;
            const float* row = e + (size_t)(j0 + rsel) * D_ + kk;
            v2f b;
            b.x = row[0];
            b.y = row[1];
            c = wmma_f32x4(afr[ks], b, c);
        }
#pragma unroll
        for (int r = 0; r < 8; ++r) {
            const int ri = r + ((lane & 16) ? 8 : 0);
            sim[ri * N_ + j0 + rsel] = c[r];
        }
    }
    __syncthreads();

    // Top-K selection: each wave owns rows {wave, wave+8}; uniform loop trip
    // counts across the block so __syncthreads() inside is legal.
    for (int rr = 0; rr < 2; ++rr) {
        const int row = wave + rr * 8;
        const int ig = i0 + row;             // global row (diag mask)
        float* srow = sim + row * N_;
        for (int t = 0; t < K_; ++t) {
            float best = NEG_BIG;
            int bj = 0x7fffffff;
            for (int j = lane; j < N_; j += 32) {
                float v = (j == ig) ? NEG_BIG : srow[j];
                if (v > best) { best = v; bj = j; }
            }
            for (int m = 16; m >= 1; m >>= 1) {
                float ov = __shfl_xor(best, m, 32);
                int   oj = __shfl_xor(bj,   m, 32);
                if (ov > best || (ov == best && oj < bj)) { best = ov; bj = oj; }
            }
            if (lane == 0) {
                tvals[(size_t)ig * K_ + t] = best;
                tidx[(size_t)ig * K_ + t]  = bj;
                srow[bj] = NEG_BIG;          // exclude for next pass
            }
            __syncthreads();                 // make srow update visible
        }
    }
}

// ---------------- K4: h = xp @ lin_w^T  (WMMA, M=BN, N=64, K=128) ----------
__global__ void h_gemm_kernel(const float* __restrict__ xp,
                              const float* __restrict__ lin_w,
                              float* __restrict__ h) {
    const int m0 = blockIdx.x * 16;
    const int lane = threadIdx.x & 31;
    const int wave = threadIdx.x >> 5;   // 4 waves -> 4 N-tiles
    const int n0 = wave * 16;
    const int rsel = lane & 15;
    const int khalf = (lane >> 4) << 1;
    v8f c = {};
#pragma unroll
    for (int ks = 0; ks < W_ / 4; ++ks) {
        const int kk = ks * 4 + khalf;
        v2f a, b;
        const float* ar = xp + (size_t)(m0 + rsel) * W_ + kk;
        a.x = ar[0]; a.y = ar[1];
        const float* br = lin_w + (size_t)(n0 + rsel) * W_ + kk;  // B[k][n] = lin_w[n][k]
        b.x = br[0]; b.y = br[1];
        c = wmma_f32x4(a, b, c);
    }
#pragma unroll
    for (int r = 0; r < 8; ++r) {
        const int m = m0 + r + ((lane & 16) ? 8 : 0);
        h[(size_t)m * D_ + n0 + rsel] = c[r];
    }
}

// ---------------- K5: per-node scalars p1,p2,aconst ------------------------
__global__ void node_scalar_kernel(const float* __restrict__ h,
                                   const float* __restrict__ emb,
                                   const float* __restrict__ att_i,
                                   const float* __restrict__ att_j,
                                   const float* __restrict__ att_em_i,
                                   const float* __restrict__ att_em_j,
                                   const float* __restrict__ att_struct,
                                   float* __restrict__ p1, float* __restrict__ p2,
                                   float* __restrict__ ac) {
    const int i = blockIdx.x * 8 + (threadIdx.x >> 5);  // one wave per node
    const int lane = threadIdx.x & 31;
    const float* hr = h + (size_t)i * D_;
    const float* er = emb + (size_t)(i & (N_ - 1)) * D_;
    float s1 = 0.0f, s2 = 0.0f, s3 = 0.0f;
#pragma unroll
    for (int d = lane; d < D_; d += 32) {
        const float hv = hr[d];
        s1 += hv * att_j[d];
        s2 += hv * att_struct[d];
        s3 += hv * att_i[d] + er[d] * (att_em_i[d] + att_em_j[d]);
    }
    for (int m = 16; m >= 1; m >>= 1) {
        s1 += __shfl_xor(s1, m, 32);
        s2 += __shfl_xor(s2, m, 32);
        s3 += __shfl_xor(s3, m, 32);
    }
    if (lane == 0) { p1[i] = s1; p2[i] = s2; ac[i] = s3; }
}

// ---------------- K6: per-target attention softmax + aggregate -------------
__global__ void aggregate_kernel(const float* __restrict__ h,
                                 const float* __restrict__ tvals,
                                 const int* __restrict__ tidx,
                                 const float* __restrict__ p1,
                                 const float* __restrict__ p2,
                                 const float* __restrict__ ac,
                                 const float* __restrict__ gate,
                                 float* __restrict__ agg) {
    const int i = blockIdx.x * 8 + (threadIdx.x >> 5);  // one wave per target
    const int lane = threadIdx.x & 31;
    const int n = i & (N_ - 1);
    const int boff = (i >> 11) << 11;                   // b * N
    const float g = gate[0];
    const float aci = ac[i];

    int   srcs[K_ + 1];
    float alph[K_ + 1];
    float cw[K_ + 1];
#pragma unroll
    for (int e2 = 0; e2 < K_; ++e2) {
        const int s = tidx[(size_t)n * K_ + e2] + boff;
        const float w = tvals[(size_t)n * K_ + e2];
        const float c = (1.0f - g) + g * w;
        float al = aci + c * p1[s] + c * w * p2[s];
        al = (al >= 0.0f) ? al : SLOPE_ * al;           // leaky_relu
        srcs[e2] = s; cw[e2] = c; alph[e2] = al;
    }
    {   // self loop: ew = 1 -> c = 1
        float al = aci + p1[i] + p2[i];
        al = (al >= 0.0f) ? al : SLOPE_ * al;
        srcs[K_] = i; cw[K_] = 1.0f; alph[K_] = al;
    }
    float mx = NEG_BIG;
#pragma unroll
    for (int e2 = 0; e2 <= K_; ++e2) mx = fmaxf(mx, alph[e2]);
    float ssum = 0.0f;
    float we[K_ + 1];
#pragma unroll
    for (int e2 = 0; e2 <= K_; ++e2) { we[e2] = __expf(alph[e2] - mx); ssum += we[e2]; }
    const float inv = 1.0f / ssum;

    float a0 = 0.0f, a1 = 0.0f;
#pragma unroll
    for (int e2 = 0; e2 <= K_; ++e2) {
        const float wgt = we[e2] * inv * cw[e2];
        const float* hr = h + (size_t)srcs[e2] * D_;
        a0 += wgt * hr[lane];
        a1 += wgt * hr[lane + 32];
    }
    agg[(size_t)i * D_ + lane]      = a0;
    agg[(size_t)i * D_ + lane + 32] = a1;
}

// ---------------- K7: out = agg @ st_w^T + bias, BN, ReLU (WMMA) -----------
__global__ void out_gemm_kernel(const float* __restrict__ agg,
                                const float* __restrict__ st_w,
                                const float* __restrict__ st_b,
                                const float* __restrict__ out_bias,
                                const float* __restrict__ bn_scale,
                                const float* __restrict__ bn_bias,
                                float* __restrict__ out) {
    const int m0 = blockIdx.x * 16;
    const int lane = threadIdx.x & 31;
    const int wave = threadIdx.x >> 5;
    const int n0 = wave * 16;
    const int rsel = lane & 15;
    const int khalf = (lane >> 4) << 1;
    v8f c = {};
#pragma unroll
    for (int ks = 0; ks < D_ / 4; ++ks) {
        const int kk = ks * 4 + khalf;
        v2f a, b;
        const float* ar = agg + (size_t)(m0 + rsel) * D_ + kk;
        a.x = ar[0]; a.y = ar[1];
        const float* br = st_w + (size_t)(n0 + rsel) * D_ + kk;   // B[k][n] = st_w[n][k]
        b.x = br[0]; b.y = br[1];
        c = wmma_f32x4(a, b, c);
    }
    const float rs = rsqrtf(1.0f + EPS_);
    const int nc = n0 + rsel;
    const float add = st_b[nc] + out_bias[nc];
    const float sc = bn_scale[nc] * rs;
    const float bb = bn_bias[nc];
#pragma unroll
    for (int r = 0; r < 8; ++r) {
        const int m = m0 + r + ((lane & 16) ? 8 : 0);
        float v = (c[r] + add) * sc + bb;
        out[(size_t)m * D_ + nc] = fmaxf(v, 0.0f);
    }
}

// ---------------- host launcher ----------------
extern "C" void kernel_launch(void* const* d_in, const int* in_sizes, int n_in,
                              void* d_out, int out_size, void* d_ws, size_t ws_size,
                              hipStream_t stream) {
    const float* x          = (const float*)d_in[0];
    const float* emb_table  = (const float*)d_in[1];
    const float* lin_w      = (const float*)d_in[2];
    const float* st_w       = (const float*)d_in[3];
    const float* st_b       = (const float*)d_in[4];
    const float* att_i      = (const float*)d_in[5];
    const float* att_j      = (const float*)d_in[6];
    const float* att_em_i   = (const float*)d_in[7];
    const float* att_em_j   = (const float*)d_in[8];
    const float* att_struct = (const float*)d_in[9];
    const float* struct_gate= (const float*)d_in[10];
    const float* out_bias   = (const float*)d_in[11];
    const float* bn_scale   = (const float*)d_in[12];
    const float* bn_bias    = (const float*)d_in[13];
    const float* tcn_w1 = (const float*)d_in[14]; const float* tcn_b1 = (const float*)d_in[15];
    const float* tcn_s1 = (const float*)d_in[16]; const float* tcn_o1 = (const float*)d_in[17];
    const float* tcn_w2 = (const float*)d_in[18]; const float* tcn_b2 = (const float*)d_in[19];
    const float* tcn_s2 = (const float*)d_in[20]; const float* tcn_o2 = (const float*)d_in[21];
    const float* tcn_w3 = (const float*)d_in[22]; const float* tcn_b3 = (const float*)d_in[23];
    const float* tcn_s3 = (const float*)d_in[24]; const float* tcn_o3 = (const float*)d_in[25];
    float* out = (float*)d_out;

    // workspace carve-up (all fp32; ~17.8 MB total)
    float* xp  = (float*)d_ws;                       // [BN, W]
    float* e   = xp  + (size_t)BN_ * W_;             // [N, D] normalized emb
    float* tv  = e   + (size_t)N_ * D_;              // [N, K] topk vals
    int*   ti  = (int*)(tv + (size_t)N_ * K_);       // [N, K] topk idx
    float* h   = (float*)(ti + (size_t)N_ * K_);     // [BN, D]
    float* p1  = h   + (size_t)BN_ * D_;             // [BN]
    float* p2  = p1  + BN_;                          // [BN]
    float* ac  = p2  + BN_;                          // [BN]
    float* agg = ac  + BN_;                          // [BN, D]

    // K1: TCN + residual
    tcn_kernel<<<BN_, W_, 0, stream>>>(x,
        tcn_w1, tcn_b1, tcn_s1, tcn_o1,
        tcn_w2, tcn_b2, tcn_s2, tcn_o2,
        tcn_w3, tcn_b3, tcn_s3, tcn_o3, xp);

    // K2: normalize embedding rows
    norm_emb_kernel<<<N_, D_, 0, stream>>>(emb_table, e);

    // K3: e·e^T via fp32 WMMA + top-K  (128 KB dynamic LDS per block)
    sim_topk_kernel<<<N_ / 16, 256, 16 * N_ * sizeof(float), stream>>>(e, tv, ti);

    // K4: h = xp @ lin_w^T via WMMA
    h_gemm_kernel<<<BN_ / 16, 128, 0, stream>>>(xp, lin_w, h);

    // K5: per-node attention scalars
    node_scalar_kernel<<<BN_ / 8, 256, 0, stream>>>(h, emb_table,
        att_i, att_j, att_em_i, att_em_j, att_struct, p1, p2, ac);

    // K6: per-target softmax + aggregation
    aggregate_kernel<<<BN_ / 8, 256, 0, stream>>>(h, tv, ti, p1, p2, ac,
                                                  struct_gate, agg);

    // K7: output GEMM + bias + BN + ReLU via WMMA
    out_gemm_kernel<<<BN_ / 16, 128, 0, stream>>>(agg, st_w, st_b, out_bias,
                                                  bn_scale, bn_bias, out);
}